// Student_16192026706523
// MI455X (gfx1250) — compile-verified
//
#include <hip/hip_runtime.h>

typedef __attribute__((ext_vector_type(2))) float v2f;
typedef __attribute__((ext_vector_type(8))) float v8f;

#define GCN_N_NODES   50000
#define GCN_N_EDGES   1600000
#define GCN_IN_FEATS  128
#define GCN_HIDDEN    128
#define GCN_CLASSES   40

// ---------------- degree + norm ----------------

__global__ void gcn_deg_kernel(const int* __restrict__ src, const int* __restrict__ dst,
                               float* __restrict__ deg_out, float* __restrict__ deg_in,
                               int n_edges) {
    int e = blockIdx.x * blockDim.x + threadIdx.x;
    if (e >= n_edges) return;
    atomicAdd(&deg_out[src[e]], 1.0f);
    atomicAdd(&deg_in[dst[e]], 1.0f);
}

__global__ void gcn_norm_kernel(float* __restrict__ deg, int n) {
    int i = blockIdx.x * blockDim.x + threadIdx.x;
    if (i < n) deg[i] = rsqrtf(fmaxf(deg[i], 1.0f));
}

// ---------------- GEMM1: H1 = (X * norm_src) @ W1   [50000x128 @ 128x128] ----------------
// One 16-row M-tile per block. 8 waves, each owns a 16-col N-tile.
// V_WMMA_F32_16X16X4_F32 lane layout (ISA 7.12.2, 32-bit A 16x4):
//   lanes 0-15: M=lane, reg0=K0 reg1=K1 ; lanes 16-31: M=lane-16, reg0=K2 reg1=K3
// B assumed symmetric with M<->N.

__global__ void __launch_bounds__(256)
gcn_gemm1_kernel(const float* __restrict__ X, const float* __restrict__ norm_src,
                 const float* __restrict__ W1, float* __restrict__ H1) {
    __shared__ float As[16][132];  // pad 128 -> 132 to spread LDS banks
    const int m0 = blockIdx.x * 16;

    for (int i = threadIdx.x; i < 16 * 128; i += 256) {
        int r = i >> 7, k = i & 127;
        int row = m0 + r;
        As[r][k] = X[(size_t)row * 128 + k] * norm_src[row];
    }
    __syncthreads();

    const int wave = threadIdx.x >> 5;
    const int lane = threadIdx.x & 31;
    const int half = lane >> 4;
    const int lr   = lane & 15;
    const int n0   = wave * 16;

    v8f acc = {0.f, 0.f, 0.f, 0.f, 0.f, 0.f, 0.f, 0.f};
#pragma unroll
    for (int kk = 0; kk < 32; ++kk) {
        const int kb = kk * 4 + 2 * half;
        v2f a, b;
        a.x = As[lr][kb];
        a.y = As[lr][kb + 1];
        b.x = W1[(size_t)kb * 128 + n0 + lr];
        b.y = W1[(size_t)(kb + 1) * 128 + n0 + lr];
        acc = __builtin_amdgcn_wmma_f32_16x16x4_f32(false, a, false, b,
                                                    (short)0, acc, false, false);
    }
#pragma unroll
    for (int r = 0; r < 8; ++r) {
        int row = m0 + r + 8 * half;
        H1[(size_t)row * 128 + n0 + lr] = acc[r];
    }
}

// ---------------- Edge aggregation, 128 feats: agg1[dst] += h1[src] ----------------
// 32 lanes per edge, float4 gather per lane, 4 f32 atomics scatter.

__global__ void __launch_bounds__(256)
gcn_agg128_kernel(const float* __restrict__ H, const int* __restrict__ src,
                  const int* __restrict__ dst, float* __restrict__ A, int n_edges) {
    int tid = blockIdx.x * blockDim.x + threadIdx.x;
    int e = tid >> 5;
    int g = tid & 31;
    if (e >= n_edges) return;
    int s = src[e], d = dst[e];
    const float4 v = *(const float4*)(H + (size_t)s * 128 + g * 4);
    float* base = A + (size_t)d * 128 + g * 4;
    atomicAdd(base + 0, v.x);
    atomicAdd(base + 1, v.y);
    atomicAdd(base + 2, v.z);
    atomicAdd(base + 3, v.w);
}

// ---------------- GEMM2: H2 = relu(agg1*norm_dst + b1) * norm_src @ W2  [.. x 40] ----------------

__global__ void __launch_bounds__(128)
gcn_gemm2_kernel(const float* __restrict__ AGG1, const float* __restrict__ norm_dst,
                 const float* __restrict__ norm_src, const float* __restrict__ b1,
                 const float* __restrict__ W2, float* __restrict__ H2) {
    __shared__ float As[16][132];
    const int m0 = blockIdx.x * 16;

    for (int i = threadIdx.x; i < 16 * 128; i += 128) {
        int r = i >> 7, k = i & 127;
        int row = m0 + r;
        float v = AGG1[(size_t)row * 128 + k] * norm_dst[row] + b1[k];
        v = fmaxf(v, 0.0f);                 // relu (layer-1 output)
        As[r][k] = v * norm_src[row];       // layer-2 source scaling
    }
    __syncthreads();

    const int wave = threadIdx.x >> 5;
    if (wave >= 3) return;                  // N=40 -> 3 tiles of 16 (padded to 48)
    const int lane = threadIdx.x & 31;
    const int half = lane >> 4;
    const int lr   = lane & 15;
    const int n0   = wave * 16;
    const int col  = n0 + lr;
    const bool ok  = col < GCN_CLASSES;

    v8f acc = {0.f, 0.f, 0.f, 0.f, 0.f, 0.f, 0.f, 0.f};
#pragma unroll
    for (int kk = 0; kk < 32; ++kk) {
        const int kb = kk * 4 + 2 * half;
        v2f a, b;
        a.x = As[lr][kb];
        a.y = As[lr][kb + 1];
        b.x = ok ? W2[(size_t)kb * GCN_CLASSES + col] : 0.0f;
        b.y = ok ? W2[(size_t)(kb + 1) * GCN_CLASSES + col] : 0.0f;
        acc = __builtin_amdgcn_wmma_f32_16x16x4_f32(false, a, false, b,
                                                    (short)0, acc, false, false);
    }
    if (ok) {
#pragma unroll
        for (int r = 0; r < 8; ++r) {
            int row = m0 + r + 8 * half;
            H2[(size_t)row * GCN_CLASSES + col] = acc[r];
        }
    }
}

// ---------------- Edge aggregation, 40 feats ----------------

__global__ void __launch_bounds__(256)
gcn_agg40_kernel(const float* __restrict__ H, const int* __restrict__ src,
                 const int* __restrict__ dst, float* __restrict__ A, int n_edges) {
    int tid = blockIdx.x * blockDim.x + threadIdx.x;
    int e = tid / GCN_CLASSES;
    int f = tid - e * GCN_CLASSES;
    if (e >= n_edges) return;
    int s = src[e], d = dst[e];
    atomicAdd(A + (size_t)d * GCN_CLASSES + f, H[(size_t)s * GCN_CLASSES + f]);
}

// ---------------- Final: out = agg2 * norm_dst + b2 ----------------

__global__ void gcn_out_kernel(const float* __restrict__ AGG2, const float* __restrict__ norm_dst,
                               const float* __restrict__ b2, float* __restrict__ out, int n) {
    int tid = blockIdx.x * blockDim.x + threadIdx.x;
    if (tid >= n) return;
    int i = tid / GCN_CLASSES;
    int f = tid - i * GCN_CLASSES;
    out[tid] = AGG2[tid] * norm_dst[i] + b2[f];
}

// ---------------- launch ----------------

extern "C" void kernel_launch(void* const* d_in, const int* in_sizes, int n_in,
                              void* d_out, int out_size, void* d_ws, size_t ws_size,
                              hipStream_t stream) {
    const float* features = (const float*)d_in[0];
    const int*   src      = (const int*)d_in[1];
    const int*   dst      = (const int*)d_in[2];
    const float* W1       = (const float*)d_in[3];
    const float* b1       = (const float*)d_in[4];
    const float* W2       = (const float*)d_in[5];
    const float* b2       = (const float*)d_in[6];
    float* out = (float*)d_out;

    // Workspace layout (floats). Zeroed region is contiguous at the front.
    float* ws       = (float*)d_ws;
    float* deg_out  = ws;                   // 50000  -> becomes norm_src in place
    float* deg_in   = ws + 50000;           // 50000  -> becomes norm_dst in place
    float* agg1     = ws + 100000;          // 50000*128 = 6,400,000
    float* agg2     = ws + 6500000;         // 50000*40  = 2,000,000
    float* h1       = ws + 8500000;         // 6,400,000
    float* h2       = ws + 14900000;        // 2,000,000   (total 16.9M floats = 67.6 MB)

    // zero degrees + both aggregation accumulators in one shot
    hipMemsetAsync(ws, 0, (size_t)8500000 * sizeof(float), stream);

    gcn_deg_kernel<<<(GCN_N_EDGES + 255) / 256, 256, 0, stream>>>(src, dst, deg_out, deg_in,
                                                                  GCN_N_EDGES);
    gcn_norm_kernel<<<(100000 + 255) / 256, 256, 0, stream>>>(ws, 100000);

    gcn_gemm1_kernel<<<GCN_N_NODES / 16, 256, 0, stream>>>(features, deg_out, W1, h1);

    {
        int total = GCN_N_EDGES * 32;  // 51.2M threads
        gcn_agg128_kernel<<<(total + 255) / 256, 256, 0, stream>>>(h1, src, dst, agg1,
                                                                   GCN_N_EDGES);
    }

    gcn_gemm2_kernel<<<GCN_N_NODES / 16, 128, 0, stream>>>(agg1, deg_in, deg_out, b1, W2, h2);

    {
        long total = (long)GCN_N_EDGES * GCN_CLASSES;  // 64M threads
        gcn_agg40_kernel<<<(int)((total + 255) / 256), 256, 0, stream>>>(h2, src, dst, agg2,
                                                                         GCN_N_EDGES);
    }

    {
        int total = GCN_N_NODES * GCN_CLASSES;  // 2M
        gcn_out_kernel<<<(total + 255) / 256, 256, 0, stream>>>(agg2, deg_in, b2, out, total);
    }
}